// PrototypeLayer_16844861735549
// MI455X (gfx1250) — compile-verified
//
#include <hip/hip_runtime.h>
#include <math.h>

// Problem constants (from reference)
#define BATCH      32
#define DIM        256      // PROTOTYPE_DIM (= GEMM K)
#define NPIX       784      // 28*28 tokens  (= GEMM M, 49 tiles of 16)
#define NPROTO     2000     // prototypes    (= GEMM N)
#define NTILES     49       // NPIX / 16
#define KSTEPS     64       // DIM / 4
#define PT_BLOCK   128      // prototypes per block (8 waves * 16)
#define PBLOCKS    16       // ceil(NPROTO / PT_BLOCK)

typedef __attribute__((ext_vector_type(2))) float v2f;
typedef __attribute__((ext_vector_type(8))) float v8f;

// ---------------------------------------------------------------------------
// Kernel 1: precompute ||x_n||^2 (per batch,token) and ||p||^2 (per prototype)
// into workspace.
// ---------------------------------------------------------------------------
__global__ void proto_sq_kernel(const float* __restrict__ x,
                                const float* __restrict__ protos,
                                float* __restrict__ xsq,
                                float* __restrict__ psq)
{
    int idx = blockIdx.x * blockDim.x + threadIdx.x;
    if (idx < BATCH * NPIX) {
        int b = idx / NPIX, n = idx % NPIX;
        const float* xb = x + (size_t)b * DIM * NPIX + n;
        float s = 0.f;
        #pragma unroll 8
        for (int d = 0; d < DIM; ++d) {
            float v = xb[(size_t)d * NPIX];
            s = fmaf(v, v, s);
        }
        xsq[idx] = s;
    } else if (idx < BATCH * NPIX + NPROTO) {
        int p = idx - BATCH * NPIX;
        const float* pr = protos + (size_t)p * DIM;
        float s = 0.f;
        #pragma unroll 8
        for (int d = 0; d < DIM; ++d) {
            float v = pr[d];
            s = fmaf(v, v, s);
        }
        psq[p] = s;
    }
}

// ---------------------------------------------------------------------------
// Kernel 2: fused GEMM (f32 WMMA 16x16x4) + per-prototype min epilogue.
// grid = (PBLOCKS, BATCH), block = 256 (8 wave32).
// B (each wave's 16 prototype rows, 64 x v2f = 128 VGPRs/lane) is held in
// registers for the whole kernel; the A tile is double-staged through
// registers+LDS so global-load latency overlaps the 64-WMMA compute.
// Steady-state inner loop: LDS reads + v_wmma only, zero global traffic.
// ---------------------------------------------------------------------------
__launch_bounds__(256)
__global__ void proto_main_kernel(const float* __restrict__ x,
                                  const float* __restrict__ protos,
                                  const float* __restrict__ xsq,
                                  const float* __restrict__ psq,
                                  float* __restrict__ out)
{
    // A tile in LDS, [d][n] layout: WMMA A fetch (lanes0-15: K, lanes16-31: K+2)
    // hits disjoint 16-bank groups -> conflict-free.
    __shared__ float As[DIM][16];   // 16 KB

    const int b     = blockIdx.y;
    const int wave  = threadIdx.x >> 5;
    const int lane  = threadIdx.x & 31;
    const int col   = lane & 15;            // matrix row (A) / column (B,C)
    const int khalf = (lane >> 4) << 1;     // 0 for lanes 0-15, 2 for 16-31
    const int p0    = blockIdx.x * PT_BLOCK + wave * 16;
    // clamp prototype index for loads (tail blocks); stores predicated later.
    const int pcol  = (p0 + col < NPROTO) ? (p0 + col) : (NPROTO - 1);

    const float* xb = x + (size_t)b * DIM * NPIX;
    const float* pp = protos + (size_t)pcol * DIM;   // this lane's B row
    const float* xs = xsq + b * NPIX;

    // ---- hoist B into registers: loop-invariant across all 49 row tiles ----
    v2f breg[KSTEPS];
    #pragma unroll
    for (int kt = 0; kt < KSTEPS; ++kt) {
        const int k0 = kt * 4 + khalf;
        breg[kt].x = pp[k0];
        breg[kt].y = pp[k0 + 1];
    }

    // A-tile staging indices: thread t moves 4x float4; d = (t>>2) + 64*i,
    // n-quad = (t&3)*4  -> 16B-aligned b128 global loads and ds stores.
    const int tn4 = (threadIdx.x & 3) << 2;
    const int td  = threadIdx.x >> 2;       // 0..63

    float runmin = 3.4e38f;                 // min_n (xsq[n] - 2*cross[n,p])
    const int mbase = (lane >> 4) << 3;     // 0 or 8 (C layout M offset)

    // prologue: fetch tile 0 into registers
    float4 stage[4];
    #pragma unroll
    for (int i = 0; i < 4; ++i) {
        int d = td + (i << 6);
        stage[i] = *(const float4*)(xb + (size_t)d * NPIX + tn4);
    }

    for (int nt = 0; nt < NTILES; ++nt) {
        const int n0 = nt * 16;

        __syncthreads();   // previous tile fully consumed -> As writable
        #pragma unroll
        for (int i = 0; i < 4; ++i) {
            int d = td + (i << 6);
            *(float4*)(&As[d][tn4]) = stage[i];
        }
        __syncthreads();   // As ready

        // prefetch next tile (clamped, uniform) -> overlaps the WMMA loop
        const int nf = (nt + 1 < NTILES) ? (nt + 1) * 16 : 0;
        #pragma unroll
        for (int i = 0; i < 4; ++i) {
            int d = td + (i << 6);
            stage[i] = *(const float4*)(xb + (size_t)d * NPIX + nf + tn4);
        }

        v8f acc = {};  // C := 0
        #pragma unroll
        for (int kt = 0; kt < KSTEPS; ++kt) {
            const int k0 = kt * 4;
            v2f a;
            // A 16x4 f32 layout: lane -> row M=col, VGPR0/1 = K+khalf, +khalf+1
            a.x = As[k0 + khalf][col];
            a.y = As[k0 + khalf + 1][col];
            acc = __builtin_amdgcn_wmma_f32_16x16x4_f32(
                      false, a, false, breg[kt], (short)0, acc, false, false);
        }

        // Epilogue: fold this row tile into the running min.
        // C/D layout: VGPR r holds M = r + mbase, N = col.
        #pragma unroll
        for (int r = 0; r < 8; ++r) {
            float v = xs[n0 + mbase + r] - 2.0f * acc[r];
            runmin = fminf(runmin, v);
        }
    }

    // Lanes l and l^16 hold the two M-halves of the same column N -> combine.
    runmin = fminf(runmin, __shfl_xor(runmin, 16, 32));

    if (lane < 16 && (p0 + col) < NPROTO) {
        const int p = p0 + col;
        float d2   = fmaxf(runmin + psq[p], 1e-12f);
        float dist = sqrtf(d2);
        // max_n similarity == similarity(min_n distance): log((d+1)/(d+1e-7))
        float sim  = logf((dist + 1.0f) / (dist + 1e-7f));
        out[(size_t)b * NPROTO + p] = sim;                                // max_similarity
        out[(size_t)BATCH * NPROTO + (size_t)b * NPROTO + p] = dist;      // min_distances_per_class
    }
}

// ---------------------------------------------------------------------------
extern "C" void kernel_launch(void* const* d_in, const int* in_sizes, int n_in,
                              void* d_out, int out_size, void* d_ws, size_t ws_size,
                              hipStream_t stream)
{
    const float* x      = (const float*)d_in[0];   // (32, 256, 28, 28)
    const float* protos = (const float*)d_in[1];   // (1, 2000, 256)
    float* xsq = (float*)d_ws;                     // 32*784 floats
    float* psq = xsq + BATCH * NPIX;               // 2000 floats
    float* out = (float*)d_out;                    // 64000 sim + 64000 dist

    const int sq_total  = BATCH * NPIX + NPROTO;
    const int sq_blocks = (sq_total + 255) / 256;
    proto_sq_kernel<<<sq_blocks, 256, 0, stream>>>(x, protos, xsq, psq);

    dim3 grid(PBLOCKS, BATCH);
    proto_main_kernel<<<grid, 256, 0, stream>>>(x, protos, xsq, psq, out);
}